// ConfusionAwareFocalLoss_2808908611737
// MI455X (gfx1250) — compile-verified
//
#include <hip/hip_runtime.h>
#include <stdint.h>

#define CC 128

constexpr float LOG2E    = 1.44269504088896340736f;
constexpr float LN2      = 0.69314718055994530942f;
constexpr float SMOOTH_U = 0.1f / 128.0f;   // SMOOTHING / num_classes
constexpr float TGT_W    = 0.9f;            // 1 - SMOOTHING

typedef __attribute__((ext_vector_type(2))) float v2f;
typedef __attribute__((ext_vector_type(8))) float v8f;

// ---- CDNA5 async global->LDS copy (ASYNCcnt path) ----
__device__ __forceinline__ void async_ld_b128(unsigned lds_off, unsigned long long gaddr) {
    asm volatile("global_load_async_to_lds_b128 %0, %1, off"
                 :: "v"(lds_off), "v"(gaddr) : "memory");
}
__device__ __forceinline__ void wait_async0() {
    asm volatile("s_wait_asynccnt 0x0" ::: "memory");
}

__global__ __launch_bounds__(256) void focal_main_kernel(
    const float* __restrict__ inp, const int* __restrict__ tgt,
    const float* __restrict__ cw,  const float* __restrict__ pm,
    float* __restrict__ partial, int rowsPerBlock)
{
    __shared__ __align__(16) float ldsE[CC * CC];   // 64 KB: penalty matrix -> excess
    const int tid  = threadIdx.x;
    const int lane = tid & 31;
    const int wave = tid >> 5;

    // ---- async fill of LDS with the 128x128 penalty matrix (64 KB) ----
    {
        unsigned ldsBase = (unsigned)(uintptr_t)(void*)&ldsE[0];
        unsigned long long g = (unsigned long long)(uintptr_t)pm;
#pragma unroll
        for (int k = 0; k < 16; ++k) {
            unsigned off = (unsigned)((k * 256 + tid) * 16);
            async_ld_b128(ldsBase + off, g + off);
        }
        wait_async0();
    }
    __syncthreads();
    // transform in place: E = max(P - 1, 0)   (diag(P)==1 -> diag(E)==0)
#pragma unroll
    for (int k = 0; k < 16; ++k) {
        int idx = (k * 256 + tid) * 4;
        float4* p4 = (float4*)&ldsE[idx];
        float4 v = *p4;
        v.x = fmaxf(v.x - 1.0f, 0.0f);
        v.y = fmaxf(v.y - 1.0f, 0.0f);
        v.z = fmaxf(v.z - 1.0f, 0.0f);
        v.w = fmaxf(v.w - 1.0f, 0.0f);
        *p4 = v;
    }
    __syncthreads();

    const int rowsPerWave = rowsPerBlock >> 3;      // 8 waves per block
    size_t row0 = (size_t)blockIdx.x * rowsPerBlock + (size_t)wave * rowsPerWave;
    float acc = 0.0f;

    for (int r = 0; r < rowsPerWave; ++r) {
        size_t row = row0 + r;
        const float4 x = ((const float4*)(inp + row * CC))[lane];   // global_load_b128
        if (r + 8 < rowsPerWave)
            __builtin_prefetch(inp + (row + 8) * CC + lane * 4, 0, 1);  // global_prefetch
        int t = __builtin_amdgcn_readfirstlane(tgt[row]);
        float w = cw[t];

        // row max (wave-wide)
        float m = fmaxf(fmaxf(x.x, x.y), fmaxf(x.z, x.w));
        m = fmaxf(m, __shfl_xor(m, 1));
        m = fmaxf(m, __shfl_xor(m, 2));
        m = fmaxf(m, __shfl_xor(m, 4));
        m = fmaxf(m, __shfl_xor(m, 8));
        m = fmaxf(m, __shfl_xor(m, 16));

        // base-2 softmax: s = (x-m)*log2(e), e = 2^s
        const float nm = -m * LOG2E;
        float sx = fmaf(x.x, LOG2E, nm), sy = fmaf(x.y, LOG2E, nm);
        float sz = fmaf(x.z, LOG2E, nm), sw = fmaf(x.w, LOG2E, nm);
        float ex = __builtin_amdgcn_exp2f(sx), ey = __builtin_amdgcn_exp2f(sy);
        float ez = __builtin_amdgcn_exp2f(sz), ew = __builtin_amdgcn_exp2f(sw);

        float S = (ex + ey) + (ez + ew);
        S += __shfl_xor(S, 1);
        S += __shfl_xor(S, 2);
        S += __shfl_xor(S, 4);
        S += __shfl_xor(S, 8);
        S += __shfl_xor(S, 16);

        const float rS  = __builtin_amdgcn_rcpf(S);
        const float L2S = __builtin_amdgcn_logf(S);     // log2(S)
        const float c1  = -L2S * LN2;                   // lp = s*ln2 - ln(S)

        float px = ex * rS, py = ey * rS, pz = ez * rS, pw = ew * rS;
        float lpx = fmaf(sx, LN2, c1), lpy = fmaf(sy, LN2, c1);
        float lpz = fmaf(sz, LN2, c1), lpw = fmaf(sw, LN2, c1);
        float ox = 1.0f - px, oy = 1.0f - py, oz = 1.0f - pz, ow = 1.0f - pw;
        float flx = ox * ox * lpx, fly = oy * oy * lpy;
        float flz = oz * oz * lpz, flw = ow * ow * lpw;
        float flsum = (flx + fly) + (flz + flw);

        // confusion penalty: dot(probs, E[t,:]) from LDS
        const float4 Ev = ((const float4*)(ldsE + (size_t)t * CC))[lane];
        float pen = fmaf(Ev.x, px, fmaf(Ev.y, py, fmaf(Ev.z, pz, Ev.w * pw)));

        // combined reduction: sum_lanes [ pen - w*u*(fw*lp) ]
        float z = fmaf(-(w * SMOOTH_U), flsum, pen);
        z += __shfl_xor(z, 1);
        z += __shfl_xor(z, 2);
        z += __shfl_xor(z, 4);
        z += __shfl_xor(z, 8);
        z += __shfl_xor(z, 16);

        // target element's fw*lp (t is wave-uniform)
        int k3 = t & 3;
        float flv = (k3 == 0) ? flx : (k3 == 1) ? fly : (k3 == 2) ? flz : flw;
        float tf = __shfl(flv, t >> 2);

        acc += z - TGT_W * w * tf;
    }

    // block reduction (deterministic, reuse LDS after all waves done reading E)
    __syncthreads();
    if (lane == 0) ldsE[wave] = acc;
    __syncthreads();
    if (tid == 0) {
        float s = 0.0f;
#pragma unroll
        for (int i = 0; i < 8; ++i) s += ldsE[i];
        partial[blockIdx.x] = s;
    }
}

// Final reduction of block partials with V_WMMA_F32_16X16X4_F32:
// B = ones(4x16)  =>  D[i][n] = C[i][n] + sum_k A[i][k]  (every column is the
// exact running row-sum). 64 partials consumed per WMMA; exact f32 math.
__global__ __launch_bounds__(32) void reduce_wmma_kernel(
    const float* __restrict__ part, float* __restrict__ out, int nb, float invN)
{
    const int lane = threadIdx.x;   // one full wave32, EXEC all ones
    v8f c = {};
    v2f b; b[0] = 1.0f; b[1] = 1.0f;
    for (int i = 0; i < nb; i += 64) {
        v2f a;
        a[0] = part[i + lane];
        a[1] = part[i + 32 + lane];
        c = __builtin_amdgcn_wmma_f32_16x16x4_f32(false, a, false, b,
                                                  (short)0, c, false, false);
    }
    // lanes 0-15 hold rows 0..7 (cols n=lane), lanes 16-31 hold rows 8..15;
    // columns are identical, so sum own 8 accs and fold across the half-wave.
    float s = ((c[0] + c[1]) + (c[2] + c[3])) + ((c[4] + c[5]) + (c[6] + c[7]));
    s += __shfl_xor(s, 16);
    if (lane == 0) out[0] = s * invN;
}

extern "C" void kernel_launch(void* const* d_in, const int* in_sizes, int n_in,
                              void* d_out, int out_size, void* d_ws, size_t ws_size,
                              hipStream_t stream) {
    const float* inp = (const float*)d_in[0];
    const int*   tgt = (const int*)d_in[1];
    const float* cw  = (const float*)d_in[2];
    const float* pm  = (const float*)d_in[3];
    float* out     = (float*)d_out;
    float* partial = (float*)d_ws;

    const int n = in_sizes[1];          // N rows (targets count) = 1048576
    const int rowsPerBlock = 256;
    const int blocks = n / rowsPerBlock;   // 4096 (multiple of 64)

    focal_main_kernel<<<blocks, 256, 0, stream>>>(inp, tgt, cw, pm, partial, rowsPerBlock);
    reduce_wmma_kernel<<<1, 32, 0, stream>>>(partial, out, blocks, 1.0f / (float)n);
}